// DIN_1262720385201
// MI455X (gfx1250) — compile-verified
//
#include <hip/hip_runtime.h>
#include <math.h>

// DIN forward for MI455X (gfx1250, wave32).
// Kernel 1: gather+mean-pool into fea[B,40] (L2-resident embedding tables,
//           float2 row loads, 16-lane shfl_xor butterfly reduction).
// Kernel 2: 3-layer MLP via V_WMMA_F32_16X16X4_F32 on 16-sample tiles.
//           All padded operand loads are branchless (clamp index + select 0)
//           so the WMMA loop has no EXEC-mask toggles.

typedef __attribute__((ext_vector_type(2))) float v2f;
typedef __attribute__((ext_vector_type(8))) float v8f;

#define BATCH 16384
#define LHIST 200
#define KDIM  10
#define FEA_W 40   // 4*K
#define H1    20
#define H1P   32   // padded LDS stride for layer-1 activations
#define H2    8
#define H2P   16   // padded LDS stride for layer-2 activations

// -------------------------------------------------------------------------
// Kernel 1: embedding gather + mean pool.
// One sample per 16-lane half-group (2 samples/wave, 16 samples/block).
// -------------------------------------------------------------------------
__global__ __launch_bounds__(256) void din_pool_kernel(
    const int* __restrict__ user_id, const int* __restrict__ movie_id,
    const int* __restrict__ movie_cate, const int* __restrict__ user_rate,
    const float* __restrict__ user_emb, const float* __restrict__ movie_emb,
    const float* __restrict__ cate_emb, float* __restrict__ fea)
{
  const int tid    = threadIdx.x;
  const int lane   = tid & 31;
  const int j      = lane & 15;          // lane within 16-lane group
  const int sub    = lane >> 4;          // which sample of the wave's pair
  const int wave   = tid >> 5;
  const int sample = blockIdx.x * 16 + wave * 2 + sub;

  float c[KDIM], d[KDIM];
#pragma unroll
  for (int k = 0; k < KDIM; ++k) { c[k] = 0.f; d[k] = 0.f; }

  const int* mc = movie_cate + (size_t)sample * LHIST;
  const int* ur = user_rate  + (size_t)sample * LHIST;

  for (int it = j; it < LHIST; it += 16) {
    const int i1 = mc[it];
    const int i2 = ur[it];               // reference routes BOTH via cate_emb
    const float2* r1 = reinterpret_cast<const float2*>(cate_emb + (size_t)i1 * KDIM);
    const float2* r2 = reinterpret_cast<const float2*>(cate_emb + (size_t)i2 * KDIM);
#pragma unroll
    for (int k = 0; k < 5; ++k) {
      float2 x = r1[k];
      float2 y = r2[k];
      c[2*k]   += x.x; c[2*k+1] += x.y;
      d[2*k]   += y.x; d[2*k+1] += y.y;
    }
  }

  // Butterfly sum across the 16-lane group (wave32: width=16 keeps it local).
#pragma unroll
  for (int off = 8; off >= 1; off >>= 1) {
#pragma unroll
    for (int k = 0; k < KDIM; ++k) {
      c[k] += __shfl_xor(c[k], off, 16);
      d[k] += __shfl_xor(d[k], off, 16);
    }
  }

  const float inv = 1.0f / (float)LHIST;
  float* frow = fea + (size_t)sample * FEA_W;
  if (j == 0) {              // pooled features
#pragma unroll
    for (int k = 0; k < KDIM; ++k) frow[20 + k] = c[k] * inv;
#pragma unroll
    for (int k = 0; k < KDIM; ++k) frow[30 + k] = d[k] * inv;
  } else if (j == 1) {       // user embedding
    const float* ue = user_emb + (size_t)user_id[sample] * KDIM;
#pragma unroll
    for (int k = 0; k < KDIM; ++k) frow[k] = ue[k];
  } else if (j == 2) {       // movie embedding
    const float* me = movie_emb + (size_t)movie_id[sample] * KDIM;
#pragma unroll
    for (int k = 0; k < KDIM; ++k) frow[10 + k] = me[k];
  }
}

// -------------------------------------------------------------------------
// Kernel 2: MLP via V_WMMA_F32_16X16X4_F32.
// One 16-sample tile per wave, 8 waves per block.
// A layout:   M = lane&15, K = vgpr + 2*(lane>>4).
// B layout:   N = lane&15, K = vgpr + 2*(lane>>4).
// C/D layout: N = lane&15, M = vgpr + 8*(lane>>4).
// -------------------------------------------------------------------------
static __device__ inline v8f wmma_f32(v2f a, v2f b, v8f c) {
  return __builtin_amdgcn_wmma_f32_16x16x4_f32(
      /*neg_a=*/false, a, /*neg_b=*/false, b,
      /*c_mod=*/(short)0, c, /*reuse_a=*/false, /*reuse_b=*/false);
}

// Branchless padded B-frag load: clamp row index, load, select 0.
static __device__ inline v2f bfrag_pad(const float* __restrict__ W,
                                       int row, int nrows, int ldw, int k) {
  const bool valid = row < nrows;
  const int  r     = valid ? row : 0;      // v_cndmask on address, no branch
  float x = W[r * ldw + k];
  float y = W[r * ldw + k + 1];
  v2f b;
  b.x = valid ? x : 0.f;                   // v_cndmask on value
  b.y = valid ? y : 0.f;
  return b;
}

__global__ __launch_bounds__(256) void din_mlp_kernel(
    const float* __restrict__ fea,
    const float* __restrict__ W1, const float* __restrict__ b1,
    const float* __restrict__ W2, const float* __restrict__ b2,
    const float* __restrict__ W3, const float* __restrict__ b3,
    float* __restrict__ out)
{
  __shared__ float h1s[8][16 * H1P];   // per-wave staging (padded stride 32)
  __shared__ float h2s[8][16 * H2P];   // padded stride 16
  __shared__ float lgs[8][16 * 2];

  const int tid   = threadIdx.x;
  const int lane  = tid & 31;
  const int wave  = tid >> 5;
  const int n     = lane & 15;
  const int h     = lane >> 4;
  const int kh    = 2 * h;
  const int tile0 = (blockIdx.x * 8 + wave) * 16;

  // ---- Layer 1: fea[16x40] @ W1^T[40x20] (N padded to 32 in two tiles) ----
  v8f acc0 = {};
  v8f acc1 = {};
  const float* arow = fea + (size_t)(tile0 + n) * FEA_W + kh;
#pragma unroll
  for (int k0 = 0; k0 < FEA_W; k0 += 4) {
    v2f a;  a.x = arow[k0];  a.y = arow[k0 + 1];
    v2f b0; b0.x = W1[n * FEA_W + k0 + kh];
            b0.y = W1[n * FEA_W + k0 + kh + 1];
    v2f bt = bfrag_pad(W1, n + 16, H1, FEA_W, k0 + kh);
    acc0 = wmma_f32(a, b0, acc0);
    acc1 = wmma_f32(a, bt, acc1);
  }
  // Bias + ReLU, store to padded LDS (columns >= 20 are never read back).
  {
    const float bia0 = b1[n];
    const int   n2   = n + 16;
    const float bia1v = b1[n2 < H1 ? n2 : 0];
    const float bia1 = (n2 < H1) ? bia1v : 0.f;
#pragma unroll
    for (int r = 0; r < 8; ++r) {
      const int m = r + 8 * h;
      float v0 = acc0[r] + bia0;
      float v1 = acc1[r] + bia1;
      h1s[wave][m * H1P + n]      = v0 > 0.f ? v0 : 0.f;
      h1s[wave][m * H1P + n + 16] = v1 > 0.f ? v1 : 0.f;  // padding cols OK
    }
  }
  __syncthreads();

  // ---- Layer 2: h1[16x20] @ W2^T[20x8] (N padded to 16) ----
  v8f acc2 = {};
#pragma unroll
  for (int k0 = 0; k0 < H1; k0 += 4) {
    v2f a;  a.x = h1s[wave][n * H1P + k0 + kh];
            a.y = h1s[wave][n * H1P + k0 + kh + 1];
    v2f bb = bfrag_pad(W2, n, H2, H1, k0 + kh);
    acc2 = wmma_f32(a, bb, acc2);
  }
  {
    const float bia = b2[n < H2 ? n : 0];
#pragma unroll
    for (int r = 0; r < 8; ++r) {
      const int m = r + 8 * h;
      float v = acc2[r] + bia;
      h2s[wave][m * H2P + n] = v > 0.f ? v : 0.f;          // cols >= 8 unused
    }
  }
  __syncthreads();

  // ---- Layer 3: h2[16x8] @ W3^T[8x2] (N padded to 16) ----
  v8f acc3 = {};
#pragma unroll
  for (int k0 = 0; k0 < H2; k0 += 4) {
    v2f a;  a.x = h2s[wave][n * H2P + k0 + kh];
            a.y = h2s[wave][n * H2P + k0 + kh + 1];
    v2f bb = bfrag_pad(W3, n, 2, H2, k0 + kh);
    acc3 = wmma_f32(a, bb, acc3);
  }
  {
    const float bia = b3[n < 2 ? n : 0];
#pragma unroll
    for (int r = 0; r < 8; ++r) {
      const int m = r + 8 * h;
      if (n < 2) lgs[wave][m * 2 + n] = acc3[r] + bia;
    }
  }
  __syncthreads();

  // Softmax over 2 logits -> p[:,1] == sigmoid(z1 - z0)
  if (lane < 16) {
    const float z0 = lgs[wave][lane * 2 + 0];
    const float z1 = lgs[wave][lane * 2 + 1];
    out[tile0 + lane] = 1.0f / (1.0f + expf(z0 - z1));
  }
}

// -------------------------------------------------------------------------
extern "C" void kernel_launch(void* const* d_in, const int* in_sizes, int n_in,
                              void* d_out, int out_size, void* d_ws, size_t ws_size,
                              hipStream_t stream) {
  const int*   user_id    = (const int*)  d_in[0];
  const int*   movie_id   = (const int*)  d_in[1];
  const int*   movie_cate = (const int*)  d_in[2];
  const int*   user_rate  = (const int*)  d_in[3];
  const float* user_emb   = (const float*)d_in[4];
  const float* movie_emb  = (const float*)d_in[5];
  const float* cate_emb   = (const float*)d_in[6];
  const float* W1         = (const float*)d_in[7];
  const float* b1         = (const float*)d_in[8];
  const float* W2         = (const float*)d_in[9];
  const float* b2         = (const float*)d_in[10];
  const float* W3         = (const float*)d_in[11];
  const float* b3         = (const float*)d_in[12];
  float*       out        = (float*)d_out;
  float*       fea        = (float*)d_ws;   // BATCH * 40 floats = 2.62 MB

  din_pool_kernel<<<BATCH / 16, 256, 0, stream>>>(
      user_id, movie_id, movie_cate, user_rate,
      user_emb, movie_emb, cate_emb, fea);

  din_mlp_kernel<<<BATCH / 128, 256, 0, stream>>>(
      fea, W1, b1, W2, b2, W3, b3, out);
}